// CustomCRFLoss_63728724738764
// MI455X (gfx1250) — compile-verified
//
#include <hip/hip_runtime.h>
#include <hip/hip_bf16.h>

#if defined(__HIP_DEVICE_COMPILE__)
#if !__has_builtin(__builtin_amdgcn_wmma_f32_16x16x4_f32)
#error "device pass: missing __builtin_amdgcn_wmma_f32_16x16x4_f32 on this toolchain"
#endif
#endif

typedef __attribute__((ext_vector_type(2))) float v2f;
typedef __attribute__((ext_vector_type(8))) float v8f;

#define NB 8
#define HW 128
#define PLANE (HW * HW)          // 16384
#define NCLS 2
#define SPATIAL_W 1.0f
#define BILATERAL_W 1.0f
#define INV_2TA2 0.5f            // 1/(2*theta_alpha^2)
#define INV_2TB2 0.5f            // 1/(2*theta_beta^2)
#define INV_TG 1.0f              // 1/theta_gamma
#define NTOT 131072.0f           // n*h*w
#define LOG2E 1.4426950408889634f

__device__ __forceinline__ float fast_exp2(float x) {
#if defined(__HIP_DEVICE_COMPILE__)
#if __has_builtin(__builtin_amdgcn_exp2f)
  return __builtin_amdgcn_exp2f(x);   // v_exp_f32
#else
  return exp2f(x);                    // ocml, also lowers to v_exp_f32
#endif
#else
  return 0.0f;
#endif
}

// ---------------------------------------------------------------------------
// 16x16 tile of exp(-scale * ||x_j - x_k||^2) for 3-channel pixel features
// taken from three contiguous 128-float channel rows, entirely on the matrix
// unit. Two chained V_WMMA_F32_16X16X4_F32:
//   C1 = A_n x B_n          with A_n=(-s*n_j, 1, 0, 0), B_n=(1, -s*n_k, 0, 0)
//   T  = A_x x B_y + C1     with A_x=(x0,x1,x2,0),      B_y=2s*(y0,y1,y2,0)
// so T[j,k] = -s*(n_j + n_k - 2 x.y) = -log2(e)*scale*||x_j-x_k||^2, and the
// result is exp2(T) elementwise: one v_exp_f32 per element, no rescale, no
// norm-broadcast shuffles. Result in WMMA C/D layout (M = v + 8*hi, N = m).
// ---------------------------------------------------------------------------
__device__ __forceinline__ v8f exp_kernel_tile(const float* __restrict__ c0,
                                               const float* __restrict__ c1,
                                               const float* __restrict__ c2,
                                               int j0, int k0, float scale) {
#if defined(__HIP_DEVICE_COMPILE__)
  const int lane = threadIdx.x & 31;
  const int m = lane & 15;
  const bool hi = lane >= 16;
  const float s = scale * LOG2E;

  // Feature operands. A 16x4: lanes 0-15 hold K=0,1 (VGPR0,1); lanes 16-31 K=2,3.
  // B 4x16: VGPR0 = K0 (lanes 0-15) / K2 (lanes 16-31), VGPR1 = K1/K3.
  v2f ax, by;
  if (!hi) {
    ax.x = c0[j0 + m];  ax.y = c1[j0 + m];
    by.x = c0[k0 + m];  by.y = c1[k0 + m];
  } else {
    ax.x = c2[j0 + m];  ax.y = 0.0f;
    by.x = c2[k0 + m];  by.y = 0.0f;
  }

  // Full squared norms: low lanes contribute c0^2+c1^2, high lanes c2^2.
  float pa = ax.x * ax.x + ax.y * ax.y;
  float pb = by.x * by.x + by.y * by.y;
  float nj = pa + __shfl_xor(pa, 16, 32);  // norm of A-column (j0+m)
  float nk = pb + __shfl_xor(pb, 16, 32);  // norm of B-column (k0+m)

  // Norm operands occupy the lanes that already hold nj/nk (low-lane K slots).
  v2f an, bn;
  an.x = hi ? 0.0f : (-s * nj);  an.y = hi ? 0.0f : 1.0f;   // K0,K1 / K2,K3
  bn.x = hi ? 0.0f : 1.0f;       bn.y = hi ? 0.0f : (-s * nk);

  // Scale B features by 2s so the chained product is -s * distance directly.
  by.x *= 2.0f * s;  by.y *= 2.0f * s;

  v8f cz = {};
  v8f c1acc = __builtin_amdgcn_wmma_f32_16x16x4_f32(false, an, false, bn,
                                                    (short)0, cz, false, false);
  v8f t = __builtin_amdgcn_wmma_f32_16x16x4_f32(false, ax, false, by,
                                                (short)0, c1acc, false, false);
  v8f r;
#pragma unroll
  for (int v = 0; v < 8; ++v) r[v] = fast_exp2(t[v]);
  return r;
#else
  return v8f{};
#endif
}

// ---------------------------------------------------------------------------
// Kernel 0: per-plane 128x128 transpose of images (24 planes) for coalesced
// column-kernel loads.
// ---------------------------------------------------------------------------
__global__ void crf_transpose(const float* __restrict__ img, float* __restrict__ imgT) {
  int idx = blockIdx.x * blockDim.x + threadIdx.x;
  if (idx >= NB * 3 * PLANE) return;
  int plane = idx >> 14;
  int rem = idx & (PLANE - 1);
  int y = rem >> 7, x = rem & 127;
  imgT[plane * PLANE + x * HW + y] = img[plane * PLANE + y * HW + x];
}

// ---------------------------------------------------------------------------
// Kernel 1: U[b][j] = sum_i unary[b,i,j], unary = logsumexp(l0,l1) - l[label].
// One thread per (b,j): column loads are coalesced across threads.
// ---------------------------------------------------------------------------
__global__ void crf_unary_colsum(const float* __restrict__ logits,
                                 const int* __restrict__ labels,
                                 float* __restrict__ U) {
  int idx = blockIdx.x * blockDim.x + threadIdx.x;
  if (idx >= NB * HW) return;
  int b = idx >> 7, j = idx & 127;
  const float* l0 = logits + b * NCLS * PLANE + j;
  const float* l1 = l0 + PLANE;
  const int* lab = labels + b * PLANE + j;
  float s = 0.0f;
  for (int i = 0; i < HW; ++i) {
    float a = l0[i * HW], c = l1[i * HW];
    float mx = fmaxf(a, c);
    float lse = mx + __logf(__expf(a - mx) + __expf(c - mx));
    float chosen = (lab[i * HW] == 0) ? a : c;
    s += lse - chosen;
  }
  U[idx] = s;
}

// ---------------------------------------------------------------------------
// Kernel 2: row-kernel term. One wave per (b, tj, tk) 16x16 tile of M[b];
// loop over all 128 image rows, chained WMMA + exp2, accumulate in registers.
// 512 waves x 256 WMMAs.
// ---------------------------------------------------------------------------
__global__ void crf_row_kernel(const float* __restrict__ img, float* __restrict__ M) {
  int waveId = blockIdx.x * (blockDim.x >> 5) + (threadIdx.x >> 5);  // 0..511
  int b = waveId >> 6;
  int tj = (waveId >> 3) & 7;
  int tk = waveId & 7;
  const float* base = img + b * 3 * PLANE;
  v8f acc = {};
  for (int i = 0; i < HW; ++i) {
    const float* c0 = base + i * HW;
    const float* c1 = base + PLANE + i * HW;
    const float* c2 = base + 2 * PLANE + i * HW;
    v8f t = exp_kernel_tile(c0, c1, c2, tj * 16, tk * 16, INV_2TA2);
#pragma unroll
    for (int v = 0; v < 8; ++v) acc[v] += t[v];
  }
  const int lane = threadIdx.x & 31;
  const int m = lane & 15;
  const int hiB = (lane >= 16) ? 8 : 0;
  float* Mb = M + b * PLANE;
#pragma unroll
  for (int v = 0; v < 8; ++v)
    Mb[(tj * 16 + hiB + v) * HW + tk * 16 + m] = SPATIAL_W * acc[v];
}

// ---------------------------------------------------------------------------
// Kernel 3: column-kernel term. One wave per (b, j): pairwise kernel among the
// 128 rows of column j (read from transposed image), column-summed over i,
// accumulated into M[b][j, k]. 1024 waves x 128 WMMAs.
// ---------------------------------------------------------------------------
__global__ void crf_col_kernel(const float* __restrict__ imgT, float* __restrict__ M) {
  int waveId = blockIdx.x * (blockDim.x >> 5) + (threadIdx.x >> 5);  // 0..1023
  int b = waveId >> 7;
  int j = waveId & 127;
  const float* base = imgT + b * 3 * PLANE;
  const float* c0 = base + j * HW;
  const float* c1 = base + PLANE + j * HW;
  const float* c2 = base + 2 * PLANE + j * HW;
  const int lane = threadIdx.x & 31;
  const int m = lane & 15;
  float* Mb = M + b * PLANE;
  for (int tk = 0; tk < 8; ++tk) {
    float s = 0.0f;
    for (int ti = 0; ti < 8; ++ti) {
      v8f t = exp_kernel_tile(c0, c1, c2, ti * 16, tk * 16, INV_2TB2);
      float local = 0.0f;
#pragma unroll
      for (int v = 0; v < 8; ++v) local += t[v];   // sums M = 8*hi + {0..7} for this N
      s += local + __shfl_xor(local, 16, 32);      // full sum over all 16 M for this N
    }
    if (lane < 16) {
      int idx = j * HW + tk * 16 + m;
      Mb[idx] = Mb[idx] + BILATERAL_W * s;
    }
  }
}

// ---------------------------------------------------------------------------
// Kernel 4: mean-field on column sums. Single block, 128 threads, fixed-order
// (deterministic) reductions. out = mean(unary) - S/(tg*N).
// ---------------------------------------------------------------------------
__global__ void crf_final(const float* __restrict__ M, const float* __restrict__ U,
                          float* __restrict__ out) {
  __shared__ float q[HW];
  __shared__ float red[HW];
  const int t = threadIdx.x;  // 0..127

  float tu = 0.0f;
  for (int b = 0; b < NB; ++b) tu += U[b * HW + t];
  red[t] = tu;
  __syncthreads();
  for (int off = 64; off >= 1; off >>= 1) {
    if (t < off) red[t] += red[t + off];
    __syncthreads();
  }
  float total_unary = red[0];
  __syncthreads();

  float S = 0.0f;  // thread-local partial of sum over iterations of y_t
  for (int b = 0; b < NB; ++b) {
    float qj = U[b * HW + t];
    const float* Mb = M + b * PLANE;
    for (int it = 0; it < 5; ++it) {
      q[t] = qj;
      __syncthreads();
      float y = 0.0f;
      for (int k = 0; k < HW; ++k) y += Mb[t * HW + k] * q[k];
      S += y;
      qj -= y * INV_TG;
      __syncthreads();
    }
  }
  red[t] = S;
  __syncthreads();
  for (int off = 64; off >= 1; off >>= 1) {
    if (t < off) red[t] += red[t + off];
    __syncthreads();
  }
  if (t == 0) out[0] = total_unary / NTOT - (red[0] * INV_TG) / NTOT;
}

// ---------------------------------------------------------------------------
extern "C" void kernel_launch(void* const* d_in, const int* in_sizes, int n_in,
                              void* d_out, int out_size, void* d_ws, size_t ws_size,
                              hipStream_t stream) {
  const float* logits = (const float*)d_in[0];  // (8,2,128,128)
  const int* labels = (const int*)d_in[1];      // (8,128,128)
  const float* images = (const float*)d_in[2];  // (8,3,128,128)
  float* out = (float*)d_out;

  float* ws = (float*)d_ws;
  float* imgT = ws;                       // 8*3*128*128 = 393216 floats
  float* M = ws + NB * 3 * PLANE;         // 8*128*128   = 131072 floats
  float* U = M + NB * PLANE;              // 8*128       = 1024 floats

  crf_transpose<<<(NB * 3 * PLANE + 255) / 256, 256, 0, stream>>>(images, imgT);
  crf_unary_colsum<<<(NB * HW + 255) / 256, 256, 0, stream>>>(logits, labels, U);
  crf_row_kernel<<<64, 256, 0, stream>>>(images, M);   // 512 waves
  crf_col_kernel<<<128, 256, 0, stream>>>(imgT, M);    // 1024 waves
  crf_final<<<1, 128, 0, stream>>>(M, U, out);
}